// AdstockTransform_56281251447315
// MI455X (gfx1250) — compile-verified
//
#include <hip/hip_runtime.h>
#include <stdint.h>

// AdstockTransform: r[t] = x[t] + sigmoid(decay[c]) * r[t-1]
// [B=64, T=8192, C=128] fp32. HBM-bandwidth-bound scan:
// 512 MiB min traffic -> ~22us floor at 23.3 TB/s.
// Single-pass blocked scan; tile staging via CDNA5 Tensor Data Mover
// (tensor_load_to_lds + s_wait_tensorcnt), 6-deep LDS pipeline.

namespace {
constexpr int kB = 64;
constexpr int kT = 8192;
constexpr int kC = 128;

constexpr int CG  = 32;          // channels per block
constexpr int NSL = 8;           // time slices per tile
constexpr int TS  = 16;          // timesteps per thread per tile
constexpr int TT  = NSL * TS;    // 128 timesteps per tile
constexpr int NTILES = kT / TT;  // 64 tiles per block
constexpr int NBUF = 6;          // LDS pipeline depth (5 tiles prefetch ahead)
constexpr int NTHREADS = CG * NSL; // 256
} // namespace

typedef unsigned int v4u __attribute__((ext_vector_type(4)));
typedef int          v4i __attribute__((ext_vector_type(4)));
typedef int          v8i __attribute__((ext_vector_type(8)));

// Issue one 2D TDM load: tile_w elements (4B each) x tile_h rows,
// row stride = stride_elems, global->LDS (contiguous rows in LDS).
__device__ __forceinline__ void tdm_load_2d(uint32_t lds_addr, uint64_t gaddr,
                                            uint32_t tile_w, uint32_t tile_h,
                                            uint32_t stride_elems)
{
    // --- D# group 0 (128b): count=1 | lds_addr | global_addr[56:0] | type=2
    v4u g0;
    g0.x = 1u;                                            // count=1, user desc
    g0.y = lds_addr;                                      // LDS byte address
    g0.z = (uint32_t)gaddr;                               // global_addr[31:0]
    g0.w = (uint32_t)((gaddr >> 32) & 0x01FFFFFFu)        // global_addr[56:32]
         | (2u << 30);                                    // type=2 ("image")

    // --- D# group 1 (256b)
    const uint32_t td0 = 1u << 20;                        // tensor_dim0 (OOB only)
    const uint32_t td1 = 1u << 20;                        // tensor_dim1 (OOB only)
    v8i g1;
    g1[0] = (int)(2u << 16);                              // wg_mask=0, data_size=2 (4B)
    g1[1] = (int)((td0 & 0xFFFFu) << 16);                 // tensor_dim0[15:0]
    g1[2] = (int)((td0 >> 16) | ((td1 & 0xFFFFu) << 16)); // td0[31:16] | td1[15:0]
    g1[3] = (int)((td1 >> 16) | (tile_w << 16));          // td1[31:16] | tile_dim0
    g1[4] = (int)tile_h;                                  // tile_dim1 | tile_dim2=0
    g1[5] = (int)stride_elems;                            // tensor_dim0_stride[31:0]
    g1[6] = 0;                                            // stride0[47:32] | stride1 lo
    g1[7] = 0;                                            // stride1 hi

    v4i g2 = {0, 0, 0, 0};                                // dims 2/3 unused (2D)
    v4i g3 = {0, 0, 0, 0};                                // dim 4 unused

#if defined(__clang_major__) && (__clang_major__ >= 23)
    v8i g4 = {0, 0, 0, 0, 0, 0, 0, 0};
    __builtin_amdgcn_tensor_load_to_lds(g0, g1, g2, g3, g4, 0);
#else
    __builtin_amdgcn_tensor_load_to_lds(g0, g1, g2, g3, 0);
#endif
}

__global__ __launch_bounds__(NTHREADS) void adstock_scan_kernel(
    const float* __restrict__ xg,
    const float* __restrict__ decay,
    float* __restrict__ out)
{
    __shared__ float xbuf[NBUF][TT * CG];   // 6 x 16 KB staging tiles
    __shared__ float carr[NSL][CG];         // per-slice local carry-outs
    __shared__ float cinArr[NSL][CG];       // per-slice corrected carry-ins
    __shared__ float runc[CG];              // running carry per channel
    __shared__ float dummy[CG];             // sink for tail dummy TDM loads

    const int tid = threadIdx.x;
    const int c   = tid & (CG - 1);         // channel within group
    const int s   = tid >> 5;               // time slice 0..7 (== wave id)
    const int bb  = blockIdx.y;             // batch row
    const int cg0 = blockIdx.x * CG;        // first channel of this block

    // d = sigmoid(decay[c]); precompute d^1..d^TS
    const float d = 1.0f / (1.0f + __expf(-decay[cg0 + c]));
    float dp[TS];
    dp[0] = d;
#pragma unroll
    for (int i = 1; i < TS; ++i) dp[i] = dp[i - 1] * d;
    const float dL = dp[TS - 1];            // d^16

    if (s == 0) runc[c] = 0.0f;

    const size_t base = (size_t)bb * kT * kC + (size_t)cg0;
    const bool issuer = (s == 0);           // wave 0 owns all TDM traffic

    auto issue = [&](int k) {
        if (!issuer) return;                // wave-uniform branch
        if (k < NTILES) {
            const int bi = k - (k / NBUF) * NBUF;
            const uint64_t ga = (uint64_t)(uintptr_t)(xg + base + (size_t)k * TT * kC);
            const uint32_t la = (uint32_t)(uintptr_t)&xbuf[bi][0];
            tdm_load_2d(la, ga, CG, TT, kC);      // 32ch x 128t tile (16 KB)
        } else {
            // tiny dummy load keeps TENSORcnt cadence constant at the tail
            const uint64_t ga = (uint64_t)(uintptr_t)xg;
            const uint32_t la = (uint32_t)(uintptr_t)&dummy[0];
            tdm_load_2d(la, ga, 4, 1, kC);        // 16 B
        }
    };

    // prologue: prefetch NBUF-1 tiles
    for (int k = 0; k < NBUF - 1; ++k) issue(k);

    for (int k = 0; k < NTILES; ++k) {
        issue(k + NBUF - 1);
        // wave 0: wait until tile k's TDM op completed (allow NBUF-1=5 in flight)
        if (issuer) __builtin_amdgcn_s_wait_tensorcnt(NBUF - 1);
        __syncthreads();                    // tile-k data visible to all waves

        const int bi = k - (k / NBUF) * NBUF;
        const int t0 = s * TS;

        // local scan with carry-in 0 (bank-conflict-free ds_load_b32)
        float r0[TS];
        float acc = 0.0f;
#pragma unroll
        for (int i = 0; i < TS; ++i) {
            acc = __builtin_fmaf(d, acc, xbuf[bi][(t0 + i) * CG + c]);
            r0[i] = acc;
        }
        carr[s][c] = acc;
        __syncthreads();                    // (also frees xbuf[bi] for reuse)

        // combine slice carries sequentially (8 fma per channel, wave 0 only)
        if (s == 0) {
            float cv = runc[c];
#pragma unroll
            for (int ss = 0; ss < NSL; ++ss) {
                cinArr[ss][c] = cv;
                cv = __builtin_fmaf(dL, cv, carr[ss][c]);
            }
            runc[c] = cv;                   // carry into next tile
        }
        __syncthreads();

        // fix up and store: r[i] = r0[i] + d^(i+1) * cin
        const float cin = cinArr[s][c];
        float* op = out + base + ((size_t)k * TT + t0) * kC + c;
#pragma unroll
        for (int i = 0; i < TS; ++i) {
            op[(size_t)i * kC] = __builtin_fmaf(dp[i], cin, r0[i]);
        }
    }
}

extern "C" void kernel_launch(void* const* d_in, const int* in_sizes, int n_in,
                              void* d_out, int out_size, void* d_ws, size_t ws_size,
                              hipStream_t stream) {
    const float* x     = (const float*)d_in[0];  // [B, T, C] fp32
    const float* decay = (const float*)d_in[1];  // [C] fp32
    float* out = (float*)d_out;                  // [B, T, C] fp32

    dim3 grid(kC / CG, kB);                      // (4, 64) = 256 blocks
    adstock_scan_kernel<<<grid, NTHREADS, 0, stream>>>(x, decay, out);
}